// GraphSage_10471130267747
// MI455X (gfx1250) — compile-verified
//
#include <hip/hip_runtime.h>
#include <hip/hip_bf16.h>

typedef float v2f __attribute__((ext_vector_type(2)));
typedef float v8f __attribute__((ext_vector_type(8)));

__device__ __forceinline__ v8f wmma4(v2f a, v2f b, v8f c) {
    // V_WMMA_F32_16X16X4_F32: (neg_a, A, neg_b, B, c_mod, C, reuse_a, reuse_b)
    return __builtin_amdgcn_wmma_f32_16x16x4_f32(false, a, false, b, (short)0, c,
                                                 false, false);
}

// ---------------------------------------------------------------- zero scratch
__global__ void zero_kernel(float4* __restrict__ p, long long nvec) {
    long long i = (long long)blockIdx.x * blockDim.x + threadIdx.x;
    if (i < nvec) p[i] = make_float4(0.f, 0.f, 0.f, 0.f);
}

// ------------------------------------------------- layer-1 GEMM: x @ [Wl | Wr]
// x: [N,128], Wl/Wr: [128,16], yl/yr: [N,16]. One wave per 16-row tile.
__global__ void gemm1_kernel(const float* __restrict__ x,
                             const float* __restrict__ Wl,
                             const float* __restrict__ Wr,
                             float* __restrict__ yl,
                             float* __restrict__ yr,
                             int ntiles) {
    int lane = threadIdx.x & 31;
    int tile = blockIdx.x * (blockDim.x >> 5) + (threadIdx.x >> 5);
    if (tile >= ntiles) return;                 // wave-uniform exit
    int col = lane & 15;
    int hi  = lane >> 4;                        // 0: K+{0,1}, 1: K+{2,3}
    int row = tile * 16 + col;                  // A-operand: lane = M
    const float* xrow = x + (size_t)row * 128;

    v8f accL = {};
    v8f accR = {};
#pragma unroll
    for (int k = 0; k < 128; k += 4) {
        int ka = k + hi * 2;
        v2f a = *(const v2f*)(xrow + ka);       // A[M=row][ka], A[M=row][ka+1]
        v2f bl, br;                             // B[K][N=col], row stride 16
        bl[0] = Wl[ka * 16 + col];
        bl[1] = Wl[(ka + 1) * 16 + col];
        br[0] = Wr[ka * 16 + col];
        br[1] = Wr[(ka + 1) * 16 + col];
        accL = wmma4(a, bl, accL);
        accR = wmma4(a, br, accR);
    }
#pragma unroll
    for (int r = 0; r < 8; r++) {               // D: VGPR r -> row r + 8*hi
        size_t m = (size_t)tile * 16 + r + hi * 8;
        yl[m * 16 + col] = accL[r];
        yr[m * 16 + col] = accR[r];
    }
}

// ------------------------------------- edge scatter of 16-dim vectors + degree
// thread = (edge, quad): accum[dst*16+q*4 .. +3] += vals[src*16+q*4 .. +3]
__global__ void scatter16_kernel(const float* __restrict__ vals,
                                 const int* __restrict__ src,
                                 const int* __restrict__ dst,
                                 float* __restrict__ accum,
                                 float* cnt, int e4) {
    int tid = blockIdx.x * blockDim.x + threadIdx.x;
    if (tid >= e4) return;
    int e = tid >> 2, q = tid & 3;
    int s = src[e], d = dst[e];
    float4 v = *(const float4*)(vals + (size_t)s * 16 + q * 4);
    float* o = accum + (size_t)d * 16 + q * 4;
    atomicAdd(o + 0, v.x);
    atomicAdd(o + 1, v.y);
    atomicAdd(o + 2, v.z);
    atomicAdd(o + 3, v.w);
    if (cnt && q == 0) atomicAdd(cnt + d, 1.0f);
}

// ------------------------------------------ h = relu(s1/max(cnt,1) + yr + b1)
__global__ void finalize1_kernel(const float* __restrict__ s1,
                                 const float* __restrict__ yr,
                                 const float* __restrict__ cnt,
                                 const float* __restrict__ b1,
                                 float* __restrict__ h, int total) {
    int tid = blockIdx.x * blockDim.x + threadIdx.x;
    if (tid >= total) return;
    int n = tid >> 4, d = tid & 15;
    float v = s1[tid] / fmaxf(cnt[n], 1.0f) + yr[tid] + b1[d];
    h[tid] = fmaxf(v, 0.0f);
}

// ------------- layer-2 fused GEMM: out = (s2/cnt) @ Wl2 + h @ Wr2 + b2, [N,40]
__global__ void gemm2_kernel(const float* __restrict__ s2,
                             const float* __restrict__ h,
                             const float* __restrict__ cnt,
                             const float* __restrict__ Wl,
                             const float* __restrict__ Wr,
                             const float* __restrict__ b2,
                             float* __restrict__ out,
                             int ntiles) {
    int lane = threadIdx.x & 31;
    int tile = blockIdx.x * (blockDim.x >> 5) + (threadIdx.x >> 5);
    if (tile >= ntiles) return;                 // wave-uniform exit
    int col = lane & 15;
    int hi  = lane >> 4;
    int row = tile * 16 + col;
    float inv = 1.0f / fmaxf(cnt[row], 1.0f);
    int c2 = 32 + col;                          // third column tile (40 cols)

    v8f acc0 = {}, acc1 = {}, acc2 = {};
#pragma unroll
    for (int k = 0; k < 16; k += 4) {           // phase 1: A = agg2, B = W_l2
        int ka = k + hi * 2;
        v2f a;
        a[0] = s2[(size_t)row * 16 + ka] * inv;
        a[1] = s2[(size_t)row * 16 + ka + 1] * inv;
        v2f b;
        b[0] = Wl[ka * 40 + col];        b[1] = Wl[(ka + 1) * 40 + col];
        acc0 = wmma4(a, b, acc0);
        b[0] = Wl[ka * 40 + 16 + col];   b[1] = Wl[(ka + 1) * 40 + 16 + col];
        acc1 = wmma4(a, b, acc1);
        b[0] = 0.f; b[1] = 0.f;
        if (c2 < 40) { b[0] = Wl[ka * 40 + c2]; b[1] = Wl[(ka + 1) * 40 + c2]; }
        acc2 = wmma4(a, b, acc2);
    }
#pragma unroll
    for (int k = 0; k < 16; k += 4) {           // phase 2: A = h, B = W_r2
        int ka = k + hi * 2;
        v2f a;
        a[0] = h[(size_t)row * 16 + ka];
        a[1] = h[(size_t)row * 16 + ka + 1];
        v2f b;
        b[0] = Wr[ka * 40 + col];        b[1] = Wr[(ka + 1) * 40 + col];
        acc0 = wmma4(a, b, acc0);
        b[0] = Wr[ka * 40 + 16 + col];   b[1] = Wr[(ka + 1) * 40 + 16 + col];
        acc1 = wmma4(a, b, acc1);
        b[0] = 0.f; b[1] = 0.f;
        if (c2 < 40) { b[0] = Wr[ka * 40 + c2]; b[1] = Wr[(ka + 1) * 40 + c2]; }
        acc2 = wmma4(a, b, acc2);
    }
#pragma unroll
    for (int r = 0; r < 8; r++) {
        size_t m = (size_t)tile * 16 + r + hi * 8;
        out[m * 40 + col]      = acc0[r] + b2[col];
        out[m * 40 + 16 + col] = acc1[r] + b2[16 + col];
        if (c2 < 40) out[m * 40 + c2] = acc2[r] + b2[c2];
    }
}

// ----------------------------------------- in-place row-wise log_softmax [N,40]
__global__ void logsoftmax_kernel(float* __restrict__ out, int nrows) {
    int lane = threadIdx.x & 31;
    int row  = blockIdx.x * (blockDim.x >> 5) + (threadIdx.x >> 5);
    if (row >= nrows) return;
    float* p = out + (size_t)row * 40;
    float v0 = p[lane];
    float v1 = (lane < 8) ? p[32 + lane] : -3.402823466e+38f;
    float m = fmaxf(v0, v1);
#pragma unroll
    for (int off = 16; off > 0; off >>= 1) m = fmaxf(m, __shfl_xor(m, off, 32));
    float s = expf(v0 - m) + ((lane < 8) ? expf(v1 - m) : 0.0f);
#pragma unroll
    for (int off = 16; off > 0; off >>= 1) s += __shfl_xor(s, off, 32);
    float lse = m + logf(s);
    p[lane] = v0 - lse;
    if (lane < 8) p[32 + lane] = v1 - lse;
}

// ------------------------------------------------------------------- launcher
extern "C" void kernel_launch(void* const* d_in, const int* in_sizes, int n_in,
                              void* d_out, int out_size, void* d_ws, size_t ws_size,
                              hipStream_t stream) {
    const float* x    = (const float*)d_in[0];
    const int*   ei   = (const int*)d_in[1];   // int32 (JAX x64 disabled)
    const float* Wl1  = (const float*)d_in[3];
    const float* b1   = (const float*)d_in[4];
    const float* Wr1  = (const float*)d_in[5];
    const float* Wl2  = (const float*)d_in[6];
    const float* b2   = (const float*)d_in[7];
    const float* Wr2  = (const float*)d_in[8];

    const int N = in_sizes[0] / 128;
    const int E = in_sizes[1] / 2;
    const int* src = ei;
    const int* dst = ei + E;

    // workspace layout (floats): cnt[N] s1[16N] s2[16N] yl[16N] yr[16N] h[16N]
    float* ws  = (float*)d_ws;
    float* cnt = ws;
    float* s1  = cnt + (size_t)N;
    float* s2  = s1 + (size_t)N * 16;
    float* yl  = s2 + (size_t)N * 16;
    float* yr  = yl + (size_t)N * 16;
    float* h   = yr + (size_t)N * 16;

    const int ntiles = (N + 15) / 16;           // 6250 row tiles
    const int e4 = E * 4;

    // 1) zero cnt + s1 + s2 (33N floats, contiguous)
    long long nvec = ((long long)N * 33) / 4;
    zero_kernel<<<(unsigned)((nvec + 255) / 256), 256, 0, stream>>>(
        (float4*)ws, nvec);

    // 2) yl = x@Wl1, yr = x@Wr1  (WMMA f32 16x16x4)
    gemm1_kernel<<<(ntiles + 7) / 8, 256, 0, stream>>>(x, Wl1, Wr1, yl, yr, ntiles);

    // 3) s1[dst] += yl[src]; cnt[dst] += 1
    scatter16_kernel<<<(e4 + 255) / 256, 256, 0, stream>>>(yl, src, dst, s1, cnt, e4);

    // 4) h = relu(s1/max(cnt,1) + yr + b1)
    finalize1_kernel<<<(N * 16 + 255) / 256, 256, 0, stream>>>(s1, yr, cnt, b1, h,
                                                               N * 16);

    // 5) s2[dst] += h[src]
    scatter16_kernel<<<(e4 + 255) / 256, 256, 0, stream>>>(h, src, dst, s2,
                                                           nullptr, e4);

    // 6) out = (s2/cnt)@Wl2 + h@Wr2 + b2   (fused WMMA)
    gemm2_kernel<<<(ntiles + 7) / 8, 256, 0, stream>>>(s2, h, cnt, Wl2, Wr2, b2,
                                                       (float*)d_out, ntiles);

    // 7) log_softmax rows in place
    logsoftmax_kernel<<<(N + 7) / 8, 256, 0, stream>>>((float*)d_out, N);
}